// TorchHawkes_52922587021743
// MI455X (gfx1250) — compile-verified
//
#include <hip/hip_runtime.h>
#include <math.h>
#include <stdint.h>

#define KDIM 512
#define NDIM 4096
#define MEMD 64

typedef float v2f __attribute__((ext_vector_type(2)));
typedef float v8f __attribute__((ext_vector_type(8)));

// ---------------------------------------------------------------------------
// Kernel 1: mu[k] = mean(obs[k,:]) / 10 + 0.01
// ---------------------------------------------------------------------------
__global__ void mu_kernel(const float* __restrict__ obs, float* __restrict__ mu) {
    const int k = blockIdx.x;
    const int t = threadIdx.x;
    __shared__ float red[256];
    float acc = 0.0f;
    for (int i = t; i < NDIM; i += 256)
        acc += obs[(size_t)k * NDIM + i];
    red[t] = acc;
    __syncthreads();
    for (int s = 128; s > 0; s >>= 1) {
        if (t < s) red[t] += red[t + s];
        __syncthreads();
    }
    if (t == 0) mu[k] = red[0] * (1.0f / (float)NDIM) * 0.1f + 0.01f;
}

// ---------------------------------------------------------------------------
// Kernel 2: fused 64-tap causal FIR filters (exact f32 replacement for the
// exponential-decay matmul and the depthwise conv).
//   S[k,i]    = sum_{m=1..64} obs[k,i-m]    * beta*exp(-m*beta)   (zero-pad)
//   lam2[k,i] = (i>=64) ? sum_{m=1..64} extobs[k,i-m] * gamma[k,64-m] : 0
// ---------------------------------------------------------------------------
__global__ void filter_kernel(const float* __restrict__ obs,
                              const float* __restrict__ extobs,
                              const float* __restrict__ gamma,
                              const float* __restrict__ beta_p,
                              float* __restrict__ S,
                              float* __restrict__ lam2) {
    const int k = blockIdx.y;        // row 0..511
    const int chunk = blockIdx.x;    // 0..15 (chunks of 256 cols)
    const int t = threadIdx.x;       // 0..255
    const float beta = beta_p[0];

    __shared__ float s_obs[320];
    __shared__ float s_ext[320];
    __shared__ float s_w1[65];       // FIR taps for S, index 1..64
    __shared__ float s_g2[65];       // reversed gamma taps, index 1..64

    const int base = chunk * 256;
    const size_t rowoff = (size_t)k * NDIM;
    for (int idx = t; idx < 320; idx += 256) {
        const int g = base - 64 + idx;
        const bool ok = (g >= 0);
        s_obs[idx] = ok ? obs[rowoff + g] : 0.0f;
        s_ext[idx] = ok ? extobs[rowoff + g] : 0.0f;
    }
    if (t < 64) {
        const int m = t + 1;
        s_w1[m] = beta * expf(-beta * (float)m);
        s_g2[m] = gamma[k * MEMD + (MEMD - m)];
    }
    __syncthreads();

    const int i  = base + t;
    const int li = 64 + t;
    float sa = 0.0f, ea = 0.0f;
#pragma unroll 8
    for (int m = 1; m <= 64; ++m) {
        sa += s_obs[li - m] * s_w1[m];
        ea += s_ext[li - m] * s_g2[m];
    }
    S[rowoff + i]    = sa;
    lam2[rowoff + i] = (i >= MEMD) ? ea : 0.0f;
}

// ---------------------------------------------------------------------------
// Kernel 3: lam1 = alpha @ S via V_WMMA_F32_16X16X4_F32.
// B-tiles (16 k-rows x 64 cols of S) are staged once per block into LDS with
// GLOBAL_LOAD_ASYNC_TO_LDS_B128 (double-buffered, ASYNCcnt-tracked), shared
// by all 8 waves -> 8x less L2 traffic than direct per-wave loads.
// Fused epilogue: lams = softplus(mu + lam1 + lam2); deterministic partial
// loglik per block.
// Block = 8 waves (tile 128M x 64N); grid = (4096/64, 512/128) = (64, 4).
// ---------------------------------------------------------------------------
#define LDS_STRIDE 72   // 16-row chunk, padded: 2*72 % 64 == 16 -> no half-wave
                        // bank overlap on the B-fragment ds_loads

__global__ void gemm_epilogue_kernel(const float* __restrict__ alpha,
                                     const float* __restrict__ S,
                                     const float* __restrict__ lam2,
                                     const float* __restrict__ mu,
                                     const float* __restrict__ obs,
                                     float* __restrict__ lams,
                                     float* __restrict__ partials) {
    __shared__ float sB[2][16 * LDS_STRIDE];   // 2 x 4.5 KB double buffer

    const int t    = threadIdx.x;
    const int lane = t & 31;
    const int wave = t >> 5;
    const int half = lane >> 4;       // 0: lanes 0-15, 1: lanes 16-31
    const int l16  = lane & 15;
    const int tileM   = blockIdx.y * 8 + wave;  // 0..31
    const int rowA    = tileM * 16 + l16;       // A-fragment row
    const int colBase = blockIdx.x * 64;        // output column base

    // Async staging coordinates: 256 threads x float4 = 16 rows x 64 cols
    const int tr = t >> 4;            // k-row within chunk, 0..15
    const int tc = (t & 15) << 2;     // col within chunk, 0,4,...,60
    const float* gsrc0 = S + (size_t)tr * NDIM + colBase + tc;
    // LDS byte offsets (ISA: LDS addr = low 32 bits of flat address)
    const uint32_t loff0 = (uint32_t)(uintptr_t)&sB[0][tr * LDS_STRIDE + tc];
    const uint32_t loff1 = (uint32_t)(uintptr_t)&sB[1][tr * LDS_STRIDE + tc];

    v8f acc[4];
#pragma unroll
    for (int j = 0; j < 4; ++j) acc[j] = (v8f){0, 0, 0, 0, 0, 0, 0, 0};

    // Issue chunk 0
    {
        const uint64_t ga = (uint64_t)(uintptr_t)gsrc0;
        asm volatile("global_load_async_to_lds_b128 %0, %1, off"
                     :: "v"(loff0), "v"(ga) : "memory");
    }

    for (int c = 0; c < KDIM / 16; ++c) {
        if (c + 1 < KDIM / 16) {   // prefetch next chunk into other buffer
            const uint64_t ga =
                (uint64_t)(uintptr_t)(gsrc0 + (size_t)(c + 1) * 16 * NDIM);
            const uint32_t lo = ((c + 1) & 1) ? loff1 : loff0;
            asm volatile("global_load_async_to_lds_b128 %0, %1, off"
                         :: "v"(lo), "v"(ga) : "memory");
            asm volatile("s_wait_asynccnt 0x1" ::: "memory"); // chunk c done
        } else {
            asm volatile("s_wait_asynccnt 0x0" ::: "memory");
        }
        __syncthreads();           // publish buffer c&1 to all waves

        const float* lB = &sB[c & 1][0];
        const int kBase = c * 16;
#pragma unroll
        for (int ks = 0; ks < 16; ks += 4) {
            const int ka = kBase + ks + 2 * half;
            // A-frag (16x4 f32): lane holds alpha[rowA][ka], alpha[rowA][ka+1]
            v2f a;
            a.x = alpha[(size_t)rowA * KDIM + ka];
            a.y = alpha[(size_t)rowA * KDIM + ka + 1];
            const int lk = ks + 2 * half;
#pragma unroll
            for (int j = 0; j < 4; ++j) {
                // B-frag (4x16 f32) from LDS
                v2f b;
                b.x = lB[lk * LDS_STRIDE + j * 16 + l16];
                b.y = lB[(lk + 1) * LDS_STRIDE + j * 16 + l16];
                acc[j] = __builtin_amdgcn_wmma_f32_16x16x4_f32(
                    false, a, false, b, (short)0, acc[j], false, false);
            }
        }
        __syncthreads();           // all waves done reading buffer c&1
    }

    // Epilogue: C/D layout — VGPR r: lanes0-15 -> M=r, lanes16-31 -> M=r+8
    const int rowBase = tileM * 16 + 8 * half;
    float partial = 0.0f;
#pragma unroll
    for (int j = 0; j < 4; ++j) {
        const int col = colBase + j * 16 + l16;
#pragma unroll
        for (int r = 0; r < 8; ++r) {
            const int row = rowBase + r;
            const size_t idx = (size_t)row * NDIM + col;
            const float val = acc[j][r] + mu[row] + lam2[idx];
            // numerically stable softplus
            const float lam = fmaxf(val, 0.0f) + log1pf(expf(-fabsf(val)));
            lams[idx] = lam;
            partial += obs[idx] * logf(lam) - lam;
        }
    }

    // Deterministic block reduction of loglik partial
    __shared__ float red[256];
    red[t] = partial;
    __syncthreads();
    for (int s = 128; s > 0; s >>= 1) {
        if (t < s) red[t] += red[t + s];
        __syncthreads();
    }
    if (t == 0)
        partials[blockIdx.y * gridDim.x + blockIdx.x] = red[0];
}

// ---------------------------------------------------------------------------
// Kernel 4: final deterministic reduction of 256 block partials -> loglik
// ---------------------------------------------------------------------------
__global__ void reduce_kernel(const float* __restrict__ partials,
                              float* __restrict__ out) {
    __shared__ float red[256];
    red[threadIdx.x] = partials[threadIdx.x];
    __syncthreads();
    for (int s = 128; s > 0; s >>= 1) {
        if (threadIdx.x < s) red[threadIdx.x] += red[threadIdx.x + s];
        __syncthreads();
    }
    if (threadIdx.x == 0) out[0] = red[0];
}

// ---------------------------------------------------------------------------
extern "C" void kernel_launch(void* const* d_in, const int* in_sizes, int n_in,
                              void* d_out, int out_size, void* d_ws, size_t ws_size,
                              hipStream_t stream) {
    const float* obs    = (const float*)d_in[0];   // (512, 4096)
    const float* extobs = (const float*)d_in[1];   // (512, 4096)
    const float* beta   = (const float*)d_in[2];   // scalar
    const float* alpha  = (const float*)d_in[3];   // (512, 512)
    const float* gamma  = (const float*)d_in[4];   // (512, 64)
    float* out = (float*)d_out;                    // [loglik, lams(512*4096)]

    char* ws = (char*)d_ws;
    float* S        = (float*)(ws);                          // 8 MB
    float* lam2     = (float*)(ws + 8388608);                // 8 MB
    float* mu       = (float*)(ws + 16777216);               // 2 KB
    float* partials = (float*)(ws + 16777216 + 4096);        // 1 KB

    mu_kernel<<<KDIM, 256, 0, stream>>>(obs, mu);
    filter_kernel<<<dim3(NDIM / 256, KDIM), 256, 0, stream>>>(
        obs, extobs, gamma, beta, S, lam2);
    gemm_epilogue_kernel<<<dim3(NDIM / 64, KDIM / 128), 256, 0, stream>>>(
        alpha, S, lam2, mu, obs, out + 1, partials);
    reduce_kernel<<<1, 256, 0, stream>>>(partials, out);
}